// HyperedgeMaxAggregator_78408922955823
// MI455X (gfx1250) — compile-verified
//
#include <hip/hip_runtime.h>

// HyperedgeMaxAggregator: out[m, :] = max over pairs e with segment_ids[e]==m
// of features[node_ids[e], :]; segment_ids sorted -> contiguous runs.
//
// Inputs (setup_inputs order):
//   d_in[0] features     float32 [100000*64]
//   d_in[1] node_ids     int     [1000000]
//   d_in[2] segment_ids  int     [1000000]  (sorted)
//   d_in[3] num_segments int     [1]
// Output: float32 [num_segments*64]

#define FEAT 64
#define LANES 32
#define BATCHES 4                      // 4 batches of 32 pairs per wave
#define WAVE_PAIRS (LANES * BATCHES)   // 128 contiguous pairs per wave
#define BLOCK 256                      // 8 waves per block (wave32)

__device__ __forceinline__ void atomic_max_f32_dev(float* addr, float val) {
    // CDNA5 native float max atomic (no return). Exact & order-independent.
    asm volatile("global_atomic_max_num_f32 %0, %1, off scope:SCOPE_DEV"
                 :: "v"(addr), "v"(val) : "memory");
}

__global__ void hyperedge_fill_neginf(float4* __restrict__ out, int n4) {
    int i = blockIdx.x * blockDim.x + threadIdx.x;
    if (i < n4) {
        const float ninf = -__builtin_inff();
        out[i] = make_float4(ninf, ninf, ninf, ninf);
    }
}

__global__ void hyperedge_segmax_kernel(const float* __restrict__ feat,
                                        const int*  __restrict__ node_ids,
                                        const int*  __restrict__ seg_ids,
                                        float*      __restrict__ out,
                                        int n_pairs) {
    const int lane = threadIdx.x & (LANES - 1);
    const int wave = blockIdx.x * (blockDim.x >> 5) + (threadIdx.x >> 5);
    const long long start = (long long)wave * WAVE_PAIRS;
    if (start >= n_pairs) return;

    const float2* __restrict__ feat2 = (const float2*)feat;

    float2 acc = make_float2(-__builtin_inff(), -__builtin_inff());
    int curSeg = -1;

    for (int b = 0; b < BATCHES; ++b) {
        const long long bs = start + (long long)b * LANES;
        if (bs >= n_pairs) break;
        const int cnt = (int)((n_pairs - bs < LANES) ? (n_pairs - bs) : LANES);

        // Coalesced id loads: one pair per lane.
        int nid = 0, sid = -1;
        const int myIdx = (int)bs + lane;
        if (lane < cnt) {
            nid = node_ids[myIdx];
            sid = seg_ids[myIdx];
            // Warm this pair's gathered feature row (256B = 2 lines).
            const char* rp = (const char*)&feat[(long long)nid * FEAT];
            __builtin_prefetch(rp,       0, 3);   // -> global_prefetch_b8
            __builtin_prefetch(rp + 128, 0, 3);
        }
        // Warm next batch's ids.
        if (b + 1 < BATCHES) {
            const long long nb = bs + LANES + lane;
            if (nb < n_pairs) {
                __builtin_prefetch(&node_ids[(int)nb], 0, 3);
                __builtin_prefetch(&seg_ids[(int)nb], 0, 3);
            }
        }

        // Walk the 32 pairs in order; running max per segment run.
        // Lane index p is wave-uniform -> v_readlane_b32 (scalar broadcast),
        // keeping the id walk on the SALU path and off the LDS pipe.
        for (int p = 0; p < cnt; ++p) {
            const int n = __builtin_amdgcn_readlane(nid, p);
            const int s = __builtin_amdgcn_readlane(sid, p);
            // Lane owns columns [2*lane, 2*lane+1] -> coalesced b64 loads.
            const float2 v = feat2[(long long)n * (FEAT / 2) + lane];
            if (s != curSeg) {                    // uniform branch across wave
                if (curSeg >= 0) {
                    float* dst = out + (long long)curSeg * FEAT + 2 * lane;
                    atomic_max_f32_dev(dst,     acc.x);
                    atomic_max_f32_dev(dst + 1, acc.y);
                }
                curSeg = s;
                acc = v;
            } else {
                acc.x = fmaxf(acc.x, v.x);
                acc.y = fmaxf(acc.y, v.y);
            }
        }
    }

    if (curSeg >= 0) {
        float* dst = out + (long long)curSeg * FEAT + 2 * lane;
        atomic_max_f32_dev(dst,     acc.x);
        atomic_max_f32_dev(dst + 1, acc.y);
    }
}

extern "C" void kernel_launch(void* const* d_in, const int* in_sizes, int n_in,
                              void* d_out, int out_size, void* d_ws, size_t ws_size,
                              hipStream_t stream) {
    const float* features = (const float*)d_in[0];
    const int*   node_ids = (const int*)d_in[1];
    const int*   seg_ids  = (const int*)d_in[2];
    float*       out      = (float*)d_out;
    const int n_pairs = in_sizes[1];

    // 1) Initialize output to -inf (segment_max identity; empty segments too).
    const int n4 = out_size / 4;  // FEAT=64 -> divisible by 4
    hipLaunchKernelGGL(hyperedge_fill_neginf,
                       dim3((n4 + BLOCK - 1) / BLOCK), dim3(BLOCK), 0, stream,
                       (float4*)out, n4);

    // 2) One wave per 128 contiguous pairs; run-compacted atomic max flushes.
    const int waves  = (n_pairs + WAVE_PAIRS - 1) / WAVE_PAIRS;
    const int wpb    = BLOCK / 32;
    const int blocks = (waves + wpb - 1) / wpb;
    hipLaunchKernelGGL(hyperedge_segmax_kernel,
                       dim3(blocks), dim3(BLOCK), 0, stream,
                       features, node_ids, seg_ids, out, n_pairs);
}